// NucleusGNN_57286273794908
// MI455X (gfx1250) — compile-verified
//
#include <hip/hip_runtime.h>

typedef __attribute__((ext_vector_type(2))) float v2f;
typedef __attribute__((ext_vector_type(8))) float v8f;

#define HID 32

// ---------------- degree / normalization ----------------

__global__ void k_deg_init(float* __restrict__ deg, int n) {
    int i = blockIdx.x * blockDim.x + threadIdx.x;
    if (i < n) deg[i] = 1.0f;                 // self-loop weight
}

__global__ void k_deg_acc(const long long* __restrict__ dst,
                          const float* __restrict__ ew,
                          float* __restrict__ deg, long long E) {
    long long e = (long long)blockIdx.x * blockDim.x + threadIdx.x;
    if (e < E) unsafeAtomicAdd(&deg[(int)dst[e]], ew[e]);
}

__global__ void k_rsqrt_inplace(float* __restrict__ d, int n) {
    int i = blockIdx.x * blockDim.x + threadIdx.x;
    if (i < n) d[i] = rsqrtf(d[i]);           // deg >= 1 always
}

// ---------------- dense GEMM via fp32 WMMA ----------------
// Y[N x 32] = X[N x K] * W[K x 32], N divisible by 16.
// One wave per 16-row tile; two 16x16 accumulators cover the 32 cols.
// A (16x4 f32): lane<16 -> M=lane, K=k0+{0,1}; lane>=16 -> M=lane-16, K=k0+{2,3}
// B (4x16 f32): lane<16 -> N=lane, K=k0+{0,1}; lane>=16 -> N=lane-16, K=k0+{2,3}
// C/D 16x16 f32: VGPR v, lane l -> M = 8*(l>>4)+v, N = l&15
template <int K>
__global__ void k_gemm_wmma(const float* __restrict__ X,
                            const float* __restrict__ W,
                            float* __restrict__ Y, int nTiles) {
    const int lane = threadIdx.x & 31;
    const int wave = threadIdx.x >> 5;
    const int tile = blockIdx.x * (blockDim.x >> 5) + wave;
    if (tile >= nTiles) return;               // wave-uniform: EXEC stays all-1s

    const int m  = lane & 15;
    const int kb = (lane >> 4) << 1;          // 0 or 2
    const float* xrow = X + (size_t)(tile * 16 + m) * K;

    v8f acc0 = {};
    v8f acc1 = {};
#pragma unroll 4
    for (int k0 = 0; k0 < K; k0 += 4) {
        v2f a = *(const v2f*)(xrow + k0 + kb);            // K=k0+kb, k0+kb+1
        const float* w0 = W + (size_t)(k0 + kb) * HID;    // row K=k0+kb
        const float* w1 = w0 + HID;                       // row K=k0+kb+1
        v2f b0, b1;
        b0.x = w0[m];       b0.y = w1[m];                 // cols 0..15
        b1.x = w0[16 + m];  b1.y = w1[16 + m];            // cols 16..31
        acc0 = __builtin_amdgcn_wmma_f32_16x16x4_f32(
            false, a, false, b0, (short)0, acc0, false, false);
        acc1 = __builtin_amdgcn_wmma_f32_16x16x4_f32(
            false, a, false, b1, (short)0, acc1, false, false);
    }

    const int crow = tile * 16 + ((lane >> 4) << 3);
    const int ccol = lane & 15;
#pragma unroll
    for (int v = 0; v < 8; ++v) {
        Y[(size_t)(crow + v) * HID + ccol]      = acc0[v];
        Y[(size_t)(crow + v) * HID + 16 + ccol] = acc1[v];
    }
}

// ---------------- self-loop init: h = dis^2 * xt ----------------

__global__ void k_self_init(const float* __restrict__ dis,
                            const float* __restrict__ xt,
                            float* __restrict__ h, long long total) {
    long long idx = (long long)blockIdx.x * blockDim.x + threadIdx.x;
    if (idx < total) {
        float d = dis[idx >> 5];
        h[idx] = d * d * xt[idx];
    }
}

// ---------------- edge scatter: one edge per wave, one feature per lane ----

__global__ void k_edge_agg(const long long* __restrict__ src,
                           const long long* __restrict__ dst,
                           const float* __restrict__ ew,
                           const float* __restrict__ dis,
                           const float* __restrict__ xt,
                           float* __restrict__ h, long long E) {
    long long e = (long long)blockIdx.x * (blockDim.x >> 5) + (threadIdx.x >> 5);
    if (e >= E) return;
    const int f = threadIdx.x & 31;
    const int s = (int)src[e];
    const int d = (int)dst[e];
    const float norm = dis[s] * ew[e] * dis[d];
    unsafeAtomicAdd(&h[(size_t)d * HID + f], norm * xt[(size_t)s * HID + f]);
}

// ---------------- bias (+optional relu) ----------------

template <bool RELU>
__global__ void k_bias_act(float* __restrict__ h, const float* __restrict__ b,
                           long long total) {
    long long idx = (long long)blockIdx.x * blockDim.x + threadIdx.x;
    if (idx < total) {
        float v = h[idx] + b[idx & 31];
        if (RELU) v = fmaxf(v, 0.0f);
        h[idx] = v;
    }
}

// ---------------- classifier head: out[i] = h[i,:] . Wc + bc ----------------

__global__ void k_head(const float* __restrict__ h, const float* __restrict__ Wc,
                       const float* __restrict__ bc, float* __restrict__ out, int n) {
    int i = blockIdx.x * blockDim.x + threadIdx.x;
    if (i >= n) return;
    const float* row = h + (size_t)i * HID;
    float s = bc[0];
#pragma unroll
    for (int f = 0; f < HID; ++f) s += row[f] * Wc[f];
    out[i] = s;
}

// ---------------- launch ----------------

extern "C" void kernel_launch(void* const* d_in, const int* in_sizes, int n_in,
                              void* d_out, int out_size, void* d_ws, size_t ws_size,
                              hipStream_t stream) {
    const float*     x   = (const float*)d_in[0];       // [N,128]
    const long long* ei  = (const long long*)d_in[1];   // [2,E] int64
    const float*     ew  = (const float*)d_in[2];       // [E]
    const float*     W1  = (const float*)d_in[3];       // [128,32]
    const float*     b1  = (const float*)d_in[4];       // [32]
    const float*     W2  = (const float*)d_in[5];       // [32,32]
    const float*     b2  = (const float*)d_in[6];       // [32]
    const float*     Wc  = (const float*)d_in[7];       // [32,1]
    const float*     bc  = (const float*)d_in[8];       // [1]
    float* out = (float*)d_out;

    const int       N = in_sizes[0] / 128;              // 100000
    const long long E = (long long)in_sizes[2];         // 3200000
    const long long* srcp = ei;
    const long long* dstp = ei + E;

    // workspace layout: dis[N] | xt[N*32] | h[N*32]
    float* dis = (float*)d_ws;
    float* xt  = dis + N;
    float* h   = xt + (size_t)N * HID;

    const long long NH = (long long)N * HID;
    const int nTiles = N / 16;                          // 6250 exact
    const dim3 blk256(256), blk128(128);
    const int gN    = (N + 255) / 256;
    const int gNH   = (int)((NH + 255) / 256);
    const int gE    = (int)((E + 255) / 256);
    const int gEw   = (int)((E + 7) / 8);               // 8 waves/block, 1 edge/wave
    const int gGemm = (nTiles + 3) / 4;                 // 4 waves/block

    // normalization: deg -> dis = rsqrt(deg)
    k_deg_init<<<gN, blk256, 0, stream>>>(dis, N);
    k_deg_acc<<<gE, blk256, 0, stream>>>(dstp, ew, dis, E);
    k_rsqrt_inplace<<<gN, blk256, 0, stream>>>(dis, N);

    // layer 1: xt = x @ W1 ; h = dis^2*xt + scatter ; relu(h + b1)
    k_gemm_wmma<128><<<gGemm, blk128, 0, stream>>>(x, W1, xt, nTiles);
    k_self_init<<<gNH, blk256, 0, stream>>>(dis, xt, h, NH);
    k_edge_agg<<<gEw, blk256, 0, stream>>>(srcp, dstp, ew, dis, xt, h, E);
    k_bias_act<true><<<gNH, blk256, 0, stream>>>(h, b1, NH);

    // layer 2: xt = h @ W2 ; h = dis^2*xt + scatter ; h + b2
    k_gemm_wmma<32><<<gGemm, blk128, 0, stream>>>(h, W2, xt, nTiles);
    k_self_init<<<gNH, blk256, 0, stream>>>(dis, xt, h, NH);
    k_edge_agg<<<gEw, blk256, 0, stream>>>(srcp, dstp, ew, dis, xt, h, E);
    k_bias_act<false><<<gNH, blk256, 0, stream>>>(h, b2, NH);

    // head
    k_head<<<gN, blk256, 0, stream>>>(h, Wc, bc, out, N);
}